// OptimizedFractalDenoise1D_62337155334624
// MI455X (gfx1250) — compile-verified
//
#include <hip/hip_runtime.h>

// OptimizedFractalDenoise1D for MI455X (gfx1250).
//
// Workload: 16 rows x 1M f32; per pass: box5/box11 reflect stencils, per-row
// std(ddof=1) of residual -> spike damp -> blend. The reference's trailing
// overlap-add (gather/scatter-add/divide-by-count) is an identity and is
// dropped. Memory-bound: ~256 MB of traffic, fits in the 192 MB L2.
//
// CDNA5 specifics used:
//  - async global->LDS staging (global_load_async_to_lds_b128/_b32, ASYNCcnt,
//    s_wait_asynccnt) for chunk+halo tiles
//  - wave32 blocks (256 threads = 8 waves)
//
// Pipeline (3 heavy launches instead of 5):
//  stats1(x)               -> acc1 (per-row sum/sumsq of x - box5(x))
//  fused(x, acc1)          -> xp (pass-1 output, in d_ws) AND acc2
//                             (pass-2 residual stats, computed from the
//                              extended xp tile entirely in LDS)
//  apply2(xp, acc2)        -> d_out
//
// Workspace: [0,128)   acc1 (16 rows x {sum,sumsq})
//            [128,256) acc2
//            [256, 256+64MB) xp intermediate

#define ROWS 16
#define LEN 1048576
#define CHUNK 2048
#define HALO 8
#define STAGE (CHUNK + 2 * HALO)      // 2064 floats staged
#define XBUF  (CHUNK + 4)             // extended xp tile: [c0-2, c0+CHUNK+2)
#define THREADS 256
#define ELEMS (CHUNK / THREADS)       // 8
#define CHUNKS_PER_ROW (LEN / CHUNK)  // 512
#define NBLOCKS (ROWS * CHUNKS_PER_ROW)

#define TREND_SCALING 0.6f
#define DETAIL_PRESERVATION 0.85f
#define SPIKE_THRESHOLD 3.5f
#define SPIKE_DAMPING 0.35f
#define EPS_F 1e-6f

__device__ __forceinline__ int mirror_idx(int i) {
  // jnp.pad 'reflect': -k -> k, (LEN-1)+k -> (LEN-1)-k
  i = (i < 0) ? -i : i;
  i = (i >= LEN) ? (2 * LEN - 2 - i) : i;
  return i;
}

// Stage STAGE floats (global row indices [g0, g0+STAGE), reflect-mirrored at
// row ends) into LDS with CDNA5 async global->LDS copies.
// Interior chunks (no reflection): b128 async loads, 4 floats/lane/op.
// Edge chunks: per-element b32 with mirrored addresses.
__device__ __forceinline__ void async_stage_row(const float* __restrict__ rowbase,
                                                int g0, float* smem) {
  const int tid = threadIdx.x;
  if (g0 >= 0 && g0 + STAGE <= LEN) {
    const float* gbase = rowbase + g0;
    for (int t = tid; t < STAGE / 4; t += THREADS) {   // 516 vec4 transfers
      const unsigned long long gaddr = (unsigned long long)(gbase + 4 * t);
      const unsigned lds_off = (unsigned)(unsigned long long)(smem + 4 * t);
      asm volatile("global_load_async_to_lds_b128 %0, %1, off"
                   :: "v"(lds_off), "v"(gaddr)
                   : "memory");
    }
  } else {
    for (int t = tid; t < STAGE; t += THREADS) {
      const int gi = mirror_idx(g0 + t);
      const unsigned long long gaddr = (unsigned long long)(rowbase + gi);
      const unsigned lds_off = (unsigned)(unsigned long long)(smem + t);
      asm volatile("global_load_async_to_lds_b32 %0, %1, off"
                   :: "v"(lds_off), "v"(gaddr)
                   : "memory");
    }
  }
  asm volatile("s_wait_asynccnt 0" ::: "memory");
  __syncthreads();
}

// One denoised sample from the staged x tile; li = smem index of the sample.
__device__ __forceinline__ float denoise_point(const float* smem, int li,
                                               float thr) {
  const float x = smem[li];
  const float s5 = smem[li - 2] + smem[li - 1] + x + smem[li + 1] + smem[li + 2];
  const float lf = 0.2f * s5;
  const float s11 = s5 + smem[li - 5] + smem[li - 4] + smem[li - 3] +
                         smem[li + 3] + smem[li + 4] + smem[li + 5];
  const float tf = s11 * (1.0f / 11.0f);
  float r = x - lf;
  r = (fabsf(r) > thr) ? r * SPIKE_DAMPING : r;
  r *= DETAIL_PRESERVATION;
  return (1.0f - TREND_SCALING) * lf + TREND_SCALING * tf + r;
}

__device__ __forceinline__ float row_threshold(const float* __restrict__ acc,
                                               int row) {
  // std with ddof=1: var = (sumsq - sum^2/N) / (N-1)
  const float sum = acc[row * 2 + 0];
  const float sq  = acc[row * 2 + 1];
  const float n   = (float)LEN;
  float var = (sq - sum * sum / n) / (n - 1.0f);
  var = var > 0.0f ? var : 0.0f;
  float sd = sqrtf(var);
  sd = sd > EPS_F ? sd : EPS_F;
  return sd * SPIKE_THRESHOLD;
}

__device__ __forceinline__ void block_reduce_acc(float s, float s2,
                                                 float* red0, float* red1,
                                                 float* __restrict__ acc,
                                                 int row) {
  const int tid = threadIdx.x;
  red0[tid] = s;
  red1[tid] = s2;
  __syncthreads();
  for (int st = THREADS / 2; st > 0; st >>= 1) {
    if (tid < st) {
      red0[tid] += red0[tid + st];
      red1[tid] += red1[tid + st];
    }
    __syncthreads();
  }
  if (tid == 0) {
    atomicAdd(&acc[row * 2 + 0], red0[0]);
    atomicAdd(&acc[row * 2 + 1], red1[0]);
  }
}

__global__ void fd_zero(float* __restrict__ acc) {
  if (threadIdx.x < ROWS * 4) acc[threadIdx.x] = 0.0f;   // acc1 + acc2
}

// Pass-1 stats: residual = x - box5(x); per-row sum / sumsq.
__global__ __launch_bounds__(THREADS)
void fd_stats(const float* __restrict__ in, float* __restrict__ acc) {
  __shared__ float smem[STAGE];
  __shared__ float red0[THREADS];
  __shared__ float red1[THREADS];

  const int bid = blockIdx.x;
  const int row = bid / CHUNKS_PER_ROW;
  const int c0  = (bid % CHUNKS_PER_ROW) * CHUNK;
  const int tid = threadIdx.x;
  const float* rowbase = in + (size_t)row * LEN;

  async_stage_row(rowbase, c0 - HALO, smem);

  float s = 0.0f, s2 = 0.0f;
#pragma unroll
  for (int e = 0; e < ELEMS; ++e) {
    const int li = HALO + tid + e * THREADS;
    const float x  = smem[li];
    const float lf = 0.2f * (smem[li - 2] + smem[li - 1] + x +
                             smem[li + 1] + smem[li + 2]);
    const float r  = x - lf;
    s  += r;
    s2 += r * r;
  }
  block_reduce_acc(s, s2, red0, red1, acc, row);
}

// Fused: pass-1 apply (x -> xp) + pass-2 residual stats.
// Computes xp on the extended range [c0-2, c0+CHUNK+2) in LDS (only needs
// x on [c0-7, c0+CHUNK+7), inside the HALO=8 stage), writes the interior to
// global, then accumulates pass-2 sum/sumsq of xp - box5(xp) from LDS.
__global__ __launch_bounds__(THREADS)
void fd_fused(const float* __restrict__ in, const float* __restrict__ accA,
              float* __restrict__ out, float* __restrict__ accB) {
  __shared__ float smem[STAGE];
  __shared__ float xbuf[XBUF];
  __shared__ float red0[THREADS];
  __shared__ float red1[THREADS];

  const int bid = blockIdx.x;
  const int row = bid / CHUNKS_PER_ROW;
  const int c0  = (bid % CHUNKS_PER_ROW) * CHUNK;
  const int tid = threadIdx.x;
  const float* rowbase = in + (size_t)row * LEN;

  async_stage_row(rowbase, c0 - HALO, smem);

  const float thr = row_threshold(accA, row);

  // Extended xp tile: element t corresponds to global pos (c0 - 2 + t),
  // whose smem index is HALO - 2 + t.  Taps reach smem[1 .. STAGE-2].
  for (int t = tid; t < XBUF; t += THREADS) {
    xbuf[t] = denoise_point(smem, HALO - 2 + t, thr);
  }
  __syncthreads();

  float* __restrict__ orow = out + (size_t)row * LEN + c0;
  float s = 0.0f, s2 = 0.0f;
#pragma unroll
  for (int e = 0; e < ELEMS; ++e) {
    const int t = 2 + tid + e * THREADS;        // xbuf index of interior elem
    const float x  = xbuf[t];
    const float lf = 0.2f * (xbuf[t - 2] + xbuf[t - 1] + x +
                             xbuf[t + 1] + xbuf[t + 2]);
    const float r  = x - lf;
    s  += r;
    s2 += r * r;
    orow[tid + e * THREADS] = x;                // pass-1 output
  }
  block_reduce_acc(s, s2, red0, red1, accB, row);
}

// Pass-2 apply: xp -> final output.
__global__ __launch_bounds__(THREADS)
void fd_apply(const float* __restrict__ in, const float* __restrict__ acc,
              float* __restrict__ out) {
  __shared__ float smem[STAGE];

  const int bid = blockIdx.x;
  const int row = bid / CHUNKS_PER_ROW;
  const int c0  = (bid % CHUNKS_PER_ROW) * CHUNK;
  const int tid = threadIdx.x;
  const float* rowbase = in + (size_t)row * LEN;

  async_stage_row(rowbase, c0 - HALO, smem);

  const float thr = row_threshold(acc, row);

  float* __restrict__ orow = out + (size_t)row * LEN + c0;
#pragma unroll
  for (int e = 0; e < ELEMS; ++e) {
    const int li = HALO + tid + e * THREADS;
    orow[tid + e * THREADS] = denoise_point(smem, li, thr);
  }
}

extern "C" void kernel_launch(void* const* d_in, const int* in_sizes, int n_in,
                              void* d_out, int out_size, void* d_ws, size_t ws_size,
                              hipStream_t stream) {
  (void)in_sizes; (void)n_in; (void)out_size; (void)ws_size;

  const float* x = (const float*)d_in[0];   // (4,4,1048576) f32
  // d_in[1]/d_in[2] are the constant 1/5 and 1/11 box kernels (folded in).
  float* out  = (float*)d_out;
  float* acc1 = (float*)d_ws;                        // 32 floats
  float* acc2 = acc1 + ROWS * 2;                     // 32 floats
  float* xp   = (float*)((char*)d_ws + 256);         // 16M floats (64 MB)

  const dim3 grid(NBLOCKS), block(THREADS);

  fd_zero <<<1, 64, 0, stream>>>(acc1);              // zeros acc1 + acc2
  fd_stats<<<grid, block, 0, stream>>>(x, acc1);     // pass-1 residual stats
  fd_fused<<<grid, block, 0, stream>>>(x, acc1, xp, acc2); // xp + pass-2 stats
  fd_apply<<<grid, block, 0, stream>>>(xp, acc2, out);     // final output
}